// SumWhileLoop_3779571220815
// MI455X (gfx1250) — compile-verified
//
#include <hip/hip_runtime.h>

typedef __attribute__((ext_vector_type(2))) float v2f;
typedef __attribute__((ext_vector_type(8))) float v8f;

#define NSEG  256
#define DCOL  128
#define PARTS 8
#define WAVES 8   // 8 waves * 16 cols = 128 columns

__device__ __forceinline__ int lower_bound_dev(const int* __restrict__ a, int n, int v) {
    int lo = 0, hi = n;
    while (lo < hi) {
        int mid = (lo + hi) >> 1;
        if (a[mid] < v) lo = mid + 1; else hi = mid;
    }
    return lo;
}

__device__ __forceinline__ v8f wmma_rowsum(v2f ones, v2f b, v8f c) {
    return __builtin_amdgcn_wmma_f32_16x16x4_f32(
        /*neg_a=*/false, ones, /*neg_b=*/false, b,
        /*c_mod=*/(short)0, c, /*reuse_a=*/false, /*reuse_b=*/false);
}

// Phase 1: per (segment, part) block, accumulate column sums of the block's
// contiguous row range using V_WMMA_F32_16X16X4_F32 with all-ones A.
// Each wave owns one 16-column tile; one WMMA consumes 4 rows x 16 cols.
// Main loop: 32 rows/iter, 16 unconditional NT b32 loads then 8 WMMAs over
// 4 independent accumulators. Guarded tail handles the last <32 rows.
__global__ void __launch_bounds__(32 * WAVES)
seg_partial_wmma(const float* __restrict__ inp,
                 const int* __restrict__ seg,
                 float* __restrict__ partials,
                 int n) {
    const int blk  = blockIdx.x;
    const int s    = blk / PARTS;
    const int part = blk % PARTS;
    const int tid  = threadIdx.x;
    const int wave = tid >> 5;
    const int lane = tid & 31;

    // Segment row range (ids are sorted) -> this part's sub-range.
    const int start = lower_bound_dev(seg, n, s);
    const int end   = lower_bound_dev(seg, n, s + 1);
    const int len   = end - start;
    const int p0 = start + (int)(((long long)len * part)       / PARTS);
    const int p1 = start + (int)(((long long)len * (part + 1)) / PARTS);

    // B-matrix lane mapping: lanes 0-15 carry rows (r, r+1), lanes 16-31 carry
    // rows (r+2, r+3); each lane holds column (16*wave + lane&15).
    const int col    = (wave << 4) + (lane & 15);
    const int rowoff = (lane >> 4) << 1;   // 0 or 2

    v8f acc0 = {}, acc1 = {}, acc2 = {}, acc3 = {};
    v2f ones; ones[0] = 1.0f; ones[1] = 1.0f;   // all-ones 16x4 A matrix

    int r = p0;
    // -------- main body: 32 rows per iteration, no bounds checks ----------
    for (; r + 32 <= p1; r += 32) {
        float lo[8], hi[8];
        #pragma unroll
        for (int u = 0; u < 8; ++u) {
            const size_t base = (size_t)(r + 4 * u + rowoff) * DCOL + col;
            lo[u] = __builtin_nontemporal_load(&inp[base]);
            hi[u] = __builtin_nontemporal_load(&inp[base + DCOL]);
        }
        #pragma unroll
        for (int u = 0; u < 8; ++u) {
            v2f b; b[0] = lo[u]; b[1] = hi[u];
            switch (u & 3) {
                case 0: acc0 = wmma_rowsum(ones, b, acc0); break;
                case 1: acc1 = wmma_rowsum(ones, b, acc1); break;
                case 2: acc2 = wmma_rowsum(ones, b, acc2); break;
                default: acc3 = wmma_rowsum(ones, b, acc3); break;
            }
        }
    }
    // -------- tail: up to 8 guarded 4-row steps ---------------------------
    for (; r < p1; r += 4) {
        const int ra = r + rowoff;
        const int rb = ra + 1;
        float b0v = (ra < p1)
            ? __builtin_nontemporal_load(&inp[(size_t)ra * DCOL + col]) : 0.0f;
        float b1v = (rb < p1)
            ? __builtin_nontemporal_load(&inp[(size_t)rb * DCOL + col]) : 0.0f;
        v2f b; b[0] = b0v; b[1] = b1v;
        acc0 = wmma_rowsum(ones, b, acc0);
    }

    // D layout: VGPR0, lanes 0-15 hold M=0, N=lane -> per-column partial sums.
    if (lane < 16) {
        partials[(size_t)blk * DCOL + col] = acc0[0] + acc1[0] + acc2[0] + acc3[0];
    }
}

// Phase 2: combine the PARTS partials per segment and divide by the count.
__global__ void __launch_bounds__(DCOL)
seg_finalize(const float* __restrict__ partials,
             const int* __restrict__ seg,
             float* __restrict__ out,
             int n) {
    const int s = blockIdx.x;
    const int c = threadIdx.x;
    const int start = lower_bound_dev(seg, n, s);
    const int end   = lower_bound_dev(seg, n, s + 1);
    const float cnt = (float)((end - start) > 1 ? (end - start) : 1);
    float sum = 0.0f;
    #pragma unroll
    for (int p = 0; p < PARTS; ++p)
        sum += partials[(size_t)(s * PARTS + p) * DCOL + c];
    out[(size_t)s * DCOL + c] = sum / cnt;
}

extern "C" void kernel_launch(void* const* d_in, const int* in_sizes, int n_in,
                              void* d_out, int out_size, void* d_ws, size_t ws_size,
                              hipStream_t stream) {
    const float* inp = (const float*)d_in[0];
    const int*   seg = (const int*)d_in[1];
    float*       out = (float*)d_out;
    float*       partials = (float*)d_ws;   // NSEG*PARTS*DCOL floats = 1 MB
    const int n = in_sizes[1];              // number of rows == len(batch_seg)

    seg_partial_wmma<<<dim3(NSEG * PARTS), dim3(32 * WAVES), 0, stream>>>(
        inp, seg, partials, n);
    seg_finalize<<<dim3(NSEG), dim3(DCOL), 0, stream>>>(partials, seg, out, n);
}